// MeshConvolution_43748536877384
// MI455X (gfx1250) — compile-verified
//
#include <hip/hip_runtime.h>
#include <math.h>

typedef __attribute__((ext_vector_type(2))) float v2f;
typedef __attribute__((ext_vector_type(8))) float v8f;

#define N_PTS    32768          // points per batch
#define B_SZ     4
#define C256     256
#define K_GEMM1  384
#define K_GEMM2  1024
#define K_GEMM3  256
#define KTILE    128
#define LSTRIDE  (KTILE + 4)    // pad: stride%64==4 -> 16 cols hit distinct banks
#define M_TILES  ((B_SZ * N_PTS) / 16)   // 8192 column tiles
#define EPS_BN   1e-5f
#define OUT_HALF (B_SZ * C256 * N_PTS)   // 33554432 floats per output tensor

// ---------------------------------------------------------------------------
// WMMA K-loop over one KTILE chunk.  A: 16x4 f32 frag (row = lane&15, K pair
// chosen by lane half).  B: 4x16 frag from LDS column-major panel.
// ---------------------------------------------------------------------------
__device__ __forceinline__ void wmma_ktile(const float* __restrict__ Wrow,
                                           const float* __restrict__ bcol,
                                           v8f& acc) {
#pragma unroll 8
  for (int k = 0; k < KTILE; k += 4) {
    v2f a = *(const v2f*)(Wrow + k);   // W[o][kc + half*2 + k .. +1]
    v2f b = *(const v2f*)(bcol + k);   // panel[col][kc_local + half*2 + k ..+1]
    acc = __builtin_amdgcn_wmma_f32_16x16x4_f32(
        /*neg_a=*/false, a, /*neg_b=*/false, b,
        /*c_mod=*/(short)0, acc, /*reuse_a=*/false, /*reuse_b=*/false);
  }
}

// ---------------------------------------------------------------------------
// Epilogue: add bias, write pre-BN activation, reduce per-channel sum / sum^2
// across the 16 columns held by the lane group, one atomicAdd per channel.
// C/D layout: VGPR v -> row M = v + 8*(lane>=16), col N = lane&15.
// ---------------------------------------------------------------------------
__device__ __forceinline__ void gemm_epilogue(v8f acc,
                                              const float* __restrict__ bias,
                                              float* __restrict__ Y,
                                              float* __restrict__ sums,
                                              float* __restrict__ sumsq,
                                              int m0, int wave, int lane) {
  const int l15  = lane & 15;
  const int half = lane >> 4;
  const int m = m0 + l15;
  const int b = m >> 15;             // N_PTS == 1<<15
  const int n = m & (N_PTS - 1);
#pragma unroll
  for (int v = 0; v < 8; ++v) {
    const int o = wave * 16 + half * 8 + v;
    float y = acc[v] + bias[o];
    Y[(b * C256 + o) * N_PTS + n] = y;
    float s = y, sq = y * y;
#pragma unroll
    for (int off = 1; off < 16; off <<= 1) {
      s  += __shfl_xor(s,  off, 32);
      sq += __shfl_xor(sq, off, 32);
    }
    if (l15 == 0) {
      atomicAdd(&sums[o],  s);
      atomicAdd(&sumsq[o], sq);
    }
  }
}

// ---------------------------------------------------------------------------
// GEMM1: y1 = Wc(256x384) * concat[spatial(64), structural(256), tex(64)] + bc
// ---------------------------------------------------------------------------
__global__ __launch_bounds__(512)
void gemm1_kernel(const float* __restrict__ spatial,
                  const float* __restrict__ structural,
                  const float* __restrict__ tex,
                  const float* __restrict__ W, const float* __restrict__ bias,
                  float* __restrict__ Y, float* __restrict__ sums,
                  float* __restrict__ sumsq) {
  __shared__ float panel[16 * LSTRIDE];
  const int tid = threadIdx.x;
  const int lane = tid & 31, wave = tid >> 5;
  const int l15 = lane & 15, half = lane >> 4;
  const int m0 = blockIdx.x * 16;
  const int m = m0 + (tid & 15);
  const int fb = m >> 15, fn = m & (N_PTS - 1);
  v8f acc = {0.f, 0.f, 0.f, 0.f, 0.f, 0.f, 0.f, 0.f};

  for (int kc = 0; kc < K_GEMM1; kc += KTILE) {
    __syncthreads();
    // Stage loads into registers first so they clause/overlap, then store.
    float vals[4];
#pragma unroll
    for (int j = 0; j < 4; ++j) {
      const int e = tid + j * 512;
      const int kg = kc + (e >> 4);
      const float* src;
      if (kg < 64)        src = spatial + (fb * 64 + kg) * N_PTS;
      else if (kg < 320)  src = structural + (fb * C256 + (kg - 64)) * N_PTS;
      else                src = tex + (fb * 64 + (kg - 320)) * N_PTS;
      vals[j] = src[fn];
    }
#pragma unroll
    for (int j = 0; j < 4; ++j) {
      const int e = tid + j * 512;
      panel[(e & 15) * LSTRIDE + (e >> 4)] = vals[j];
    }
    __syncthreads();
    const float* Wrow = W + (wave * 16 + l15) * K_GEMM1 + kc + half * 2;
    if (kc + KTILE < K_GEMM1) __builtin_prefetch(Wrow + KTILE, 0, 3);
    wmma_ktile(Wrow, panel + l15 * LSTRIDE + half * 2, acc);
  }
  gemm_epilogue(acc, bias, Y, sums, sumsq, m0, wave, lane);
}

// ---------------------------------------------------------------------------
// GEMM2: y2 = Wcat(256x1024) * cat[structural, sum, dif, div] + bcat
// The neighbor gather + derived features are fused into the panel build;
// structural is L2-resident (134 MB < 192 MB L2) so gathers never hit HBM.
// ---------------------------------------------------------------------------
__global__ __launch_bounds__(512)
void gemm2_kernel(const float* __restrict__ structural,
                  const int* __restrict__ nbr,
                  const float* __restrict__ W, const float* __restrict__ bias,
                  float* __restrict__ Y, float* __restrict__ sums,
                  float* __restrict__ sumsq) {
  __shared__ float panel[16 * LSTRIDE];
  __shared__ int sidx[16 * 3];
  const int tid = threadIdx.x;
  const int lane = tid & 31, wave = tid >> 5;
  const int l15 = lane & 15, half = lane >> 4;
  const int m0 = blockIdx.x * 16;
  const int m = m0 + (tid & 15);
  const int fb = m >> 15, fn = m & (N_PTS - 1);
  v8f acc = {0.f, 0.f, 0.f, 0.f, 0.f, 0.f, 0.f, 0.f};

  if (tid < 48) {                       // cache the 3 neighbor ids per column
    const int col = tid / 3, j = tid - col * 3;
    const int mm = m0 + col;
    const int b = mm >> 15, n = mm & (N_PTS - 1);
    sidx[col * 3 + j] = nbr[(b * N_PTS + n) * 3 + j];
  }

  for (int kc = 0; kc < K_GEMM2; kc += KTILE) {
    __syncthreads();                    // first pass also fences sidx
    const int g = kc >> 8;              // group constant per 128-chunk
    float vals[4];
#pragma unroll
    for (int j = 0; j < 4; ++j) {
      const int e = tid + j * 512;
      const int col = e & 15;
      const int c = (kc + (e >> 4)) & 255;
      const float* srow = structural + (fb * C256 + c) * N_PTS;
      if (g == 0) {
        vals[j] = srow[fn];
      } else {
        const float n0 = srow[sidx[col * 3 + 0]];
        const float n1 = srow[sidx[col * 3 + 1]];
        const float n2 = srow[sidx[col * 3 + 2]];
        if (g == 1)       vals[j] = n0 + n1 + n2;
        else if (g == 2)  vals[j] = fabsf(n2 - n1) + 2.0f * fabsf(n1 - n0);
        else {
          const float s = srow[fn];
          vals[j] = fabsf(n0 - s) + fabsf(n1 - s) + fabsf(n2 - s);
        }
      }
    }
#pragma unroll
    for (int j = 0; j < 4; ++j) {
      const int e = tid + j * 512;
      panel[(e & 15) * LSTRIDE + (e >> 4)] = vals[j];
    }
    __syncthreads();
    const float* Wrow = W + (wave * 16 + l15) * K_GEMM2 + kc + half * 2;
    if (kc + KTILE < K_GEMM2) __builtin_prefetch(Wrow + KTILE, 0, 3);
    wmma_ktile(Wrow, panel + l15 * LSTRIDE + half * 2, acc);
  }
  gemm_epilogue(acc, bias, Y, sums, sumsq, m0, wave, lane);
}

// ---------------------------------------------------------------------------
// GEMM3: y3 = Wa(256x256) * relu(BN2(y2)) + ba.  Stage-2 BN+ReLU is fused
// into the panel load (scale/shift precomputed), saving a 268 MB round trip.
// ---------------------------------------------------------------------------
__global__ __launch_bounds__(512)
void gemm3_kernel(const float* __restrict__ y2,
                  const float* __restrict__ scale2,
                  const float* __restrict__ shift2,
                  const float* __restrict__ W, const float* __restrict__ bias,
                  float* __restrict__ Y, float* __restrict__ sums,
                  float* __restrict__ sumsq) {
  __shared__ float panel[16 * LSTRIDE];
  const int tid = threadIdx.x;
  const int lane = tid & 31, wave = tid >> 5;
  const int l15 = lane & 15, half = lane >> 4;
  const int m0 = blockIdx.x * 16;
  const int m = m0 + (tid & 15);
  const int fb = m >> 15, fn = m & (N_PTS - 1);
  v8f acc = {0.f, 0.f, 0.f, 0.f, 0.f, 0.f, 0.f, 0.f};

  for (int kc = 0; kc < K_GEMM3; kc += KTILE) {
    __syncthreads();
    float vals[4], sc[4], sh[4];
#pragma unroll
    for (int j = 0; j < 4; ++j) {
      const int e = tid + j * 512;
      const int kg = kc + (e >> 4);
      vals[j] = y2[(fb * C256 + kg) * N_PTS + fn];
      sc[j] = scale2[kg];
      sh[j] = shift2[kg];
    }
#pragma unroll
    for (int j = 0; j < 4; ++j) {
      const int e = tid + j * 512;
      panel[(e & 15) * LSTRIDE + (e >> 4)] =
          fmaxf(vals[j] * sc[j] + sh[j], 0.f);
    }
    __syncthreads();
    const float* Wrow = W + (wave * 16 + l15) * K_GEMM3 + kc + half * 2;
    if (kc + KTILE < K_GEMM3) {
      __builtin_prefetch(Wrow + KTILE, 0, 3);
      __builtin_prefetch(&y2[(fb * C256 + kc + KTILE + (tid >> 4)) * N_PTS + fn], 0, 3);
    }
    wmma_ktile(Wrow, panel + l15 * LSTRIDE + half * 2, acc);
  }
  gemm_epilogue(acc, bias, Y, sums, sumsq, m0, wave, lane);
}

// ---------------------------------------------------------------------------
// Fold (sum, sum^2) into per-channel (scale, shift):  BN(x) = x*scale + shift
// ---------------------------------------------------------------------------
__global__ void finalize_stats_kernel(const float* __restrict__ sums,
                                      const float* __restrict__ sumsq,
                                      const float* __restrict__ gamma,
                                      const float* __restrict__ beta,
                                      float* __restrict__ scale,
                                      float* __restrict__ shift,
                                      float inv_cnt) {
  const int c = threadIdx.x;
  const float mean = sums[c] * inv_cnt;
  const float var  = sumsq[c] * inv_cnt - mean * mean;
  const float sc   = gamma[c] * rsqrtf(var + EPS_BN);
  scale[c] = sc;
  shift[c] = beta[c] - mean * sc;
}

// In-place BN+ReLU on a [B,256,N] tensor, float4 vectorized along N.
__global__ __launch_bounds__(256)
void bn_apply_kernel(float* __restrict__ Y,
                     const float* __restrict__ scale,
                     const float* __restrict__ shift) {
  const int i = blockIdx.x * 256 + threadIdx.x;       // float4 index
  const int c = (i >> 13) & 255;                      // (i*4 / N) % 256
  float4 y = ((float4*)Y)[i];
  const float sc = scale[c], sh = shift[c];
  y.x = fmaxf(y.x * sc + sh, 0.f);
  y.y = fmaxf(y.y * sc + sh, 0.f);
  y.z = fmaxf(y.z * sc + sh, 0.f);
  y.w = fmaxf(y.w * sc + sh, 0.f);
  ((float4*)Y)[i] = y;
}

__global__ void zero_kernel(float* __restrict__ p, int n) {
  const int i = blockIdx.x * 256 + threadIdx.x;
  if (i < n) p[i] = 0.f;
}

// ---------------------------------------------------------------------------
extern "C" void kernel_launch(void* const* d_in, const int* in_sizes, int n_in,
                              void* d_out, int out_size, void* d_ws, size_t ws_size,
                              hipStream_t stream) {
  const float* spatial    = (const float*)d_in[0];
  const float* structural = (const float*)d_in[1];
  const float* tex        = (const float*)d_in[2];
  const int*   nbr        = (const int*)d_in[3];
  const float* Wc    = (const float*)d_in[4];
  const float* bc    = (const float*)d_in[5];
  const float* gc    = (const float*)d_in[6];
  const float* betac = (const float*)d_in[7];
  const float* Wcat    = (const float*)d_in[8];
  const float* bcat    = (const float*)d_in[9];
  const float* gcat    = (const float*)d_in[10];
  const float* betacat = (const float*)d_in[11];
  const float* Wa    = (const float*)d_in[12];
  const float* ba    = (const float*)d_in[13];
  const float* ga    = (const float*)d_in[14];
  const float* betaa = (const float*)d_in[15];

  float* out = (float*)d_out;
  float* y1  = out;                  // spatial_out (pre-BN, then in-place BN)
  float* y3  = out + OUT_HALF;       // structural_out
  float* y2  = (float*)d_ws;         // stage-2 pre-BN intermediate (134 MB)
  float* st  = (float*)d_ws + OUT_HALF;   // stats block
  float* sums1 = st,        *sumsq1 = st + 256;
  float* sums2 = st + 512,  *sumsq2 = st + 768;
  float* sums3 = st + 1024, *sumsq3 = st + 1280;
  float* scale1 = st + 1536, *shift1 = st + 1792;
  float* scale2 = st + 2048, *shift2 = st + 2304;
  float* scale3 = st + 2560, *shift3 = st + 2816;

  const float inv_cnt = 1.0f / (float)(B_SZ * N_PTS);
  const int bn_blocks = (B_SZ * C256 * N_PTS / 4) / 256;   // 32768

  zero_kernel<<<6, 256, 0, stream>>>(st, 1536);

  gemm1_kernel<<<M_TILES, 512, 0, stream>>>(spatial, structural, tex, Wc, bc,
                                            y1, sums1, sumsq1);
  finalize_stats_kernel<<<1, 256, 0, stream>>>(sums1, sumsq1, gc, betac,
                                               scale1, shift1, inv_cnt);
  bn_apply_kernel<<<bn_blocks, 256, 0, stream>>>(y1, scale1, shift1);

  gemm2_kernel<<<M_TILES, 512, 0, stream>>>(structural, nbr, Wcat, bcat,
                                            y2, sums2, sumsq2);
  finalize_stats_kernel<<<1, 256, 0, stream>>>(sums2, sumsq2, gcat, betacat,
                                               scale2, shift2, inv_cnt);

  gemm3_kernel<<<M_TILES, 512, 0, stream>>>(y2, scale2, shift2, Wa, ba,
                                            y3, sums3, sumsq3);
  finalize_stats_kernel<<<1, 256, 0, stream>>>(sums3, sumsq3, ga, betaa,
                                               scale3, shift3, inv_cnt);
  bn_apply_kernel<<<bn_blocks, 256, 0, stream>>>(y3, scale3, shift3);
}